// GIB_Layer_38886633898578
// MI455X (gfx1250) — compile-verified
//
#include <hip/hip_runtime.h>
#include <hip/hip_bf16.h>

typedef __attribute__((ext_vector_type(2))) float v2f;
typedef __attribute__((ext_vector_type(8))) float v8f;

#define BB   2
#define NN   65536
#define MM   8192
#define KSUP 32
#define EPSV 1e-8f

// 128 threads = 4 waves (wave32). 16 queries per block, 8 k-lanes per query.
__global__ __launch_bounds__(128) void gib_layer_kernel(
    const float* __restrict__ points,    // (B,N,3)
    const float* __restrict__ q_coords,  // (B,M,3)
    const int*   __restrict__ sidx,      // (B,M,K)
    const float* __restrict__ cy_p,      // (16,2)
    const float* __restrict__ cone_p,    // (16,2)
    const float* __restrict__ disk_p,    // (16,2)
    const float* __restrict__ ellip_p,   // (16,3)
    const float* __restrict__ lambdas,   // (64,16)
    float* __restrict__ out)             // (B,M,16)
{
    __shared__ float sLam[64 * 16];
    __shared__ float sQ[16][65];            // 16x64 A-tile, padded: bank=(row+col)%64
    __shared__ float sCyR2[16], sCyI[16];
    __shared__ float sCoC[16],  sCoI[16];
    __shared__ float sDkA[16],  sDkB[16];
    __shared__ float sElX[16],  sElY[16], sElZ[16];

    const int tid = threadIdx.x;

    // ---- Stage 0: precompute per-GIB constants into LDS (one rcp per block, not per lane)
    if (tid < 16) {
        float a = cy_p[2 * tid], b = cy_p[2 * tid + 1];
        sCyR2[tid] = a * a;
        sCyI[tid]  = 1.0f / (2.0f * b * b + EPSV);
    } else if (tid < 32) {
        int i = tid - 16;
        float a = cone_p[2 * i], b = cone_p[2 * i + 1];
        sCoC[i] = a;
        sCoI[i] = 1.0f / (2.0f * b * b + EPSV);
    } else if (tid < 48) {
        int i = tid - 32;
        float a = disk_p[2 * i], b = disk_p[2 * i + 1];
        sDkA[i] = 1.0f / (a * a + EPSV);
        sDkB[i] = 1.0f / (b * b + EPSV);
    } else if (tid < 64) {
        int i = tid - 48;
        float a = ellip_p[3 * i], b = ellip_p[3 * i + 1], c = ellip_p[3 * i + 2];
        sElX[i] = 1.0f / (a * a + EPSV);
        sElY[i] = 1.0f / (b * b + EPSV);
        sElZ[i] = 1.0f / (c * c + EPSV);
    }
    for (int i = tid; i < 64 * 16; i += 128) sLam[i] = lambdas[i];
    __syncthreads();

    // ---- Stage 1: transcendental-bound GIB evaluation
    const int  ql = tid >> 3;                       // query within tile, 0..15
    const int  k0 = tid & 7;                        // k-lane, 0..7
    const long qg = (long)blockIdx.x * 16 + ql;     // flat (b*M+m), 0..16383
    const int  b  = (int)(qg >> 13);                // / M (M=8192)

    const float qx = q_coords[qg * 3 + 0];
    const float qy = q_coords[qg * 3 + 1];
    const float qz = q_coords[qg * 3 + 2];

    float rzv[4], r2v[4], rx2[4], ry2[4], rz2[4], sqv[4];
    #pragma unroll
    for (int j = 0; j < 4; ++j) {
        int  k   = k0 + 8 * j;
        int  idx = sidx[qg * KSUP + k];
        long po  = ((long)b * NN + idx) * 3;
        float rx = points[po + 0] - qx;
        float ry = points[po + 1] - qy;
        float rz = points[po + 2] - qz;
        rzv[j] = rz;
        rx2[j] = rx * rx;
        ry2[j] = ry * ry;
        rz2[j] = rz * rz;
        r2v[j] = rx2[j] + ry2[j];
        sqv[j] = sqrtf(r2v[j] + EPSV);
    }

    const float scale = 1.0f / 32.0f;

    // cylinders -> g in [0,16)
    #pragma unroll 4
    for (int g = 0; g < 16; ++g) {
        float r0 = sCyR2[g], inv = sCyI[g], s = 0.0f;
        #pragma unroll
        for (int j = 0; j < 4; ++j) { float t = r2v[j] - r0; s += __expf(-t * t * inv); }
        s += __shfl_xor(s, 1); s += __shfl_xor(s, 2); s += __shfl_xor(s, 4);
        if (k0 == 0) sQ[ql][g] = s * scale;
    }
    // cones -> g in [16,32)
    #pragma unroll 4
    for (int g = 0; g < 16; ++g) {
        float c0 = sCoC[g], inv = sCoI[g], s = 0.0f;
        #pragma unroll
        for (int j = 0; j < 4; ++j) { float t = sqv[j] - c0 * rzv[j]; s += __expf(-t * t * inv); }
        s += __shfl_xor(s, 1); s += __shfl_xor(s, 2); s += __shfl_xor(s, 4);
        if (k0 == 0) sQ[ql][16 + g] = s * scale;
    }
    // disks -> g in [32,48)
    #pragma unroll 4
    for (int g = 0; g < 16; ++g) {
        float ia = sDkA[g], ib = sDkB[g], s = 0.0f;
        #pragma unroll
        for (int j = 0; j < 4; ++j) { s += __expf(-(r2v[j] * ia + rz2[j] * ib)); }
        s += __shfl_xor(s, 1); s += __shfl_xor(s, 2); s += __shfl_xor(s, 4);
        if (k0 == 0) sQ[ql][32 + g] = s * scale;
    }
    // ellipsoids -> g in [48,64)
    #pragma unroll 4
    for (int g = 0; g < 16; ++g) {
        float ix = sElX[g], iy = sElY[g], iz = sElZ[g], s = 0.0f;
        #pragma unroll
        for (int j = 0; j < 4; ++j) { s += __expf(-(rx2[j] * ix + ry2[j] * iy + rz2[j] * iz)); }
        s += __shfl_xor(s, 1); s += __shfl_xor(s, 2); s += __shfl_xor(s, 4);
        if (k0 == 0) sQ[ql][48 + g] = s * scale;
    }
    __syncthreads();

    // ---- Stage 2: project q_out (16x64) @ lambdas (64x16) with f32 WMMA, wave 0 only.
    // V_WMMA_F32_16X16X4_F32 layouts (ISA 7.12.2):
    //   A 16x4:  lane<16 -> A[L][K=0],A[L][K=1]; lane>=16 -> A[L-16][K=2],A[L-16][K=3]
    //   B 4x16:  VGPR0 lanes0-15 = row K=0, lanes16-31 = row K=2; VGPR1 = K=1 / K=3
    //   D 16x16: VGPR j: lanes0-15 -> M=j, lanes16-31 -> M=j+8
    if (tid < 32) {
        const int hv  = tid >> 4;      // 0: K=0,1 half | 1: K=2,3 half
        const int l16 = tid & 15;
        v8f acc = {0.f, 0.f, 0.f, 0.f, 0.f, 0.f, 0.f, 0.f};
        #pragma unroll
        for (int kk = 0; kk < 16; ++kk) {
            const int kb = 4 * kk + 2 * hv;
            v2f av, bv;
            av.x = sQ[l16][kb + 0];
            av.y = sQ[l16][kb + 1];
            bv.x = sLam[(kb + 0) * 16 + l16];
            bv.y = sLam[(kb + 1) * 16 + l16];
            acc = __builtin_amdgcn_wmma_f32_16x16x4_f32(
                /*neg_a=*/false, av, /*neg_b=*/false, bv,
                /*c_mod=*/(short)0, acc, /*reuse_a=*/false, /*reuse_b=*/false);
        }
        const long obase = (long)blockIdx.x * 16;
        #pragma unroll
        for (int j = 0; j < 8; ++j) {
            const int row = j + 8 * hv;
            out[(obase + row) * 16 + l16] = acc[j];
        }
    }
}

extern "C" void kernel_launch(void* const* d_in, const int* in_sizes, int n_in,
                              void* d_out, int out_size, void* d_ws, size_t ws_size,
                              hipStream_t stream) {
    const float* points = (const float*)d_in[0];
    const float* qc     = (const float*)d_in[1];
    const int*   sidx   = (const int*)d_in[2];
    // d_in[3] = mc_points : unused by the reference computation
    const float* cy     = (const float*)d_in[4];
    const float* cone   = (const float*)d_in[5];
    const float* disk   = (const float*)d_in[6];
    const float* ellip  = (const float*)d_in[7];
    const float* lam    = (const float*)d_in[8];
    float* out = (float*)d_out;

    const int tiles = (BB * MM) / 16;   // 1024 blocks, 4 waves each
    gib_layer_kernel<<<tiles, 128, 0, stream>>>(points, qc, sidx, cy, cone, disk, ellip, lam, out);
}